// GCNModel_61933428417025
// MI455X (gfx1250) — compile-verified
//
#include <hip/hip_runtime.h>

// ---------------------------------------------------------------------------
// GCN (100k nodes, 1.6M edges) — memory/atomic-bound; GEMMs on f32 WMMA.
// Pipeline: deg -> dinv -> [h1 = x@W1 (WMMA)] -> edge scatter (f32 atomics)
//           -> fused selfloop+bias+relu -> [h2 = x1@W2 (WMMA)] -> scatter
//           -> fused selfloop+bias+relu+fc reduction.
// ---------------------------------------------------------------------------

typedef __attribute__((ext_vector_type(2))) float v2f;
typedef __attribute__((ext_vector_type(8))) float v8f;

#define GCN_N 100000

// deg[col[e]] += 1 over the 1.6M real edges (self-loop +1 folded into dinv).
__global__ void deg_kernel(const int* __restrict__ col, float* __restrict__ deg, int E) {
    int e = blockIdx.x * blockDim.x + threadIdx.x;
    if (e < E) atomicAdd(&deg[col[e]], 1.0f);
}

// dinv[i] = rsqrt(deg[i] + 1)   (deg >= 1 always, so no zero guard needed)
__global__ void dinv_kernel(float* __restrict__ deg, int n) {
    int i = blockIdx.x * blockDim.x + threadIdx.x;
    if (i < n) deg[i] = rsqrtf(deg[i] + 1.0f);
}

// H[n,32] = X[n,16] @ W[16,32].  One 16-row tile per wave; two 16x16 f32
// accumulators (cols 0-15, 16-31); K=16 as 4 chained V_WMMA_F32_16X16X4_F32.
__global__ void gemm_k16_n32_wmma(const float* __restrict__ X, const float* __restrict__ W,
                                  float* __restrict__ H, int n) {
    int wave   = blockIdx.x * (blockDim.x >> 5) + (threadIdx.x >> 5);
    int nTiles = (n + 15) >> 4;
    if (wave >= nTiles) return;                 // wave-uniform: EXEC stays all-1s
    int lane = threadIdx.x & 31;
    int mi   = lane & 15;                       // A row / B,C,D column
    int kb   = (lane >> 4) << 1;                // K sub-offset: 0 (lanes 0-15) or 2
    int r0   = wave << 4;
    bool full = (r0 + 16 <= n);                 // wave-uniform
    int rld  = full ? (r0 + mi) : min(r0 + mi, n - 1);
    const float* xr = X + (size_t)rld * 16;

    v8f acc0 = {}; v8f acc1 = {};
#pragma unroll
    for (int kk = 0; kk < 16; kk += 4) {
        v2f a = *(const v2f*)(xr + kk + kb);    // 8B aligned: row stride 64B, kk+kb even
        v2f b0, b1;
        b0.x = W[(kk + kb) * 32 + mi];          b0.y = W[(kk + kb + 1) * 32 + mi];
        b1.x = W[(kk + kb) * 32 + mi + 16];     b1.y = W[(kk + kb + 1) * 32 + mi + 16];
        acc0 = __builtin_amdgcn_wmma_f32_16x16x4_f32(false, a, false, b0, (short)0, acc0, false, false);
        acc1 = __builtin_amdgcn_wmma_f32_16x16x4_f32(false, a, false, b1, (short)0, acc1, false, false);
    }
    int mh = (lane >> 4) << 3;                  // C/D: VGPR v holds row v + 8*(lane>=16)
    float* hb = H + (size_t)(r0 + mh) * 32 + mi;
    if (full) {                                 // straight-line store clause (imm offsets)
#pragma unroll
        for (int v = 0; v < 8; v++) {
            hb[v * 32]      = acc0[v];
            hb[v * 32 + 16] = acc1[v];
        }
    } else {
#pragma unroll
        for (int v = 0; v < 8; v++) {
            if (r0 + mh + v < n) {
                hb[v * 32]      = acc0[v];
                hb[v * 32 + 16] = acc1[v];
            }
        }
    }
}

// H[n,16] = X[n,32] @ W[32,16].  K=32 as 8 chained f32 WMMAs, one accumulator.
__global__ void gemm_k32_n16_wmma(const float* __restrict__ X, const float* __restrict__ W,
                                  float* __restrict__ H, int n) {
    int wave   = blockIdx.x * (blockDim.x >> 5) + (threadIdx.x >> 5);
    int nTiles = (n + 15) >> 4;
    if (wave >= nTiles) return;
    int lane = threadIdx.x & 31;
    int mi   = lane & 15;
    int kb   = (lane >> 4) << 1;
    int r0   = wave << 4;
    bool full = (r0 + 16 <= n);                 // wave-uniform
    int rld  = full ? (r0 + mi) : min(r0 + mi, n - 1);
    const float* xr = X + (size_t)rld * 32;

    v8f acc = {};
#pragma unroll
    for (int kk = 0; kk < 32; kk += 4) {
        v2f a = *(const v2f*)(xr + kk + kb);
        v2f b;
        b.x = W[(kk + kb) * 16 + mi];
        b.y = W[(kk + kb + 1) * 16 + mi];
        acc = __builtin_amdgcn_wmma_f32_16x16x4_f32(false, a, false, b, (short)0, acc, false, false);
    }
    int mh = (lane >> 4) << 3;
    float* hb = H + (size_t)(r0 + mh) * 16 + mi;
    if (full) {
#pragma unroll
        for (int v = 0; v < 8; v++) hb[v * 16] = acc[v];
    } else {
#pragma unroll
        for (int v = 0; v < 8; v++)
            if (r0 + mh + v < n) hb[v * 16] = acc[v];
    }
}

// Edge message passing: agg[col[e]][j] += h[row[e]][j] * dinv[row]*dinv[col].
// C=32: one edge per wave (lane==column -> coalesced 128B gather + 128B atomic).
// C=16: two edges per wave.
template <int C>
__global__ void scatter_kernel(const int* __restrict__ row, const int* __restrict__ col,
                               const float* __restrict__ h, const float* __restrict__ dinv,
                               float* __restrict__ agg, int E) {
    long long t = (long long)blockIdx.x * blockDim.x + threadIdx.x;
    int e = (int)(t / C);
    int j = (int)(t % C);
    if (e < E) {
        int r = row[e], c = col[e];
        float nrm = dinv[r] * dinv[c];
        atomicAdd(&agg[(size_t)c * C + j], h[(size_t)r * C + j] * nrm);
    }
}

// Fused epilogue: agg += selfloop (h*dinv^2) + bias, then ReLU (in place).
__global__ void selfloop_bias_relu(float* __restrict__ agg, const float* __restrict__ h,
                                   const float* __restrict__ dinv, const float* __restrict__ b,
                                   int n, int C) {
    long long t = (long long)blockIdx.x * blockDim.x + threadIdx.x;
    int i = (int)(t / C);
    int j = (int)(t % C);
    if (i < n) {
        float di = dinv[i];
        float v  = agg[(size_t)i * C + j] + h[(size_t)i * C + j] * di * di + b[j];
        agg[(size_t)i * C + j] = v > 0.0f ? v : 0.0f;
    }
}

// Final: out[i] = sum_j relu(agg2[i][j] + h2[i][j]*dinv^2 + b2[j]) * fcW[j] + fcb.
__global__ void final_kernel(const float* __restrict__ agg2, const float* __restrict__ h2,
                             const float* __restrict__ dinv, const float* __restrict__ b2,
                             const float* __restrict__ fcW, const float* __restrict__ fcb,
                             float* __restrict__ out, int n) {
    int i = blockIdx.x * blockDim.x + threadIdx.x;
    if (i < n) {
        float d2 = dinv[i] * dinv[i];
        float s = 0.0f;
#pragma unroll
        for (int j = 0; j < 16; j++) {
            float v = agg2[(size_t)i * 16 + j] + h2[(size_t)i * 16 + j] * d2 + b2[j];
            v = v > 0.0f ? v : 0.0f;
            s += v * fcW[j];
        }
        out[i] = s + fcb[0];
    }
}

extern "C" void kernel_launch(void* const* d_in, const int* in_sizes, int n_in,
                              void* d_out, int out_size, void* d_ws, size_t ws_size,
                              hipStream_t stream) {
    const int*   edge = (const int*)d_in[0];   // [2,E] int32 row-major
    const float* x0   = (const float*)d_in[1]; // [N,16]
    const float* W1   = (const float*)d_in[2]; // [16,32]
    const float* b1   = (const float*)d_in[3]; // [32]
    const float* W2   = (const float*)d_in[4]; // [32,16]
    const float* b2   = (const float*)d_in[5]; // [16]
    const float* fcW  = (const float*)d_in[6]; // [16,1]
    const float* fcb  = (const float*)d_in[7]; // [1]
    float*       out  = (float*)d_out;         // [N,1]

    const int n = GCN_N;
    const int E = in_sizes[0] / 2;
    const int* row = edge;
    const int* col = edge + E;

    // Workspace layout (floats): dinv | h1[n*32] | agg1[n*32] | h2[n*16] | agg2[n*16]
    float* ws   = (float*)d_ws;
    size_t npad = ((size_t)n + 63) & ~(size_t)63;
    float* dinv = ws;
    float* h1   = dinv + npad;
    float* agg1 = h1   + (size_t)n * 32;
    float* h2   = agg1 + (size_t)n * 32;
    float* agg2 = h2   + (size_t)n * 16;

    hipMemsetAsync(dinv, 0, (size_t)n * sizeof(float), stream);
    hipMemsetAsync(agg1, 0, (size_t)n * 32 * sizeof(float), stream);
    hipMemsetAsync(agg2, 0, (size_t)n * 16 * sizeof(float), stream);

    deg_kernel <<<(E + 255) / 256, 256, 0, stream>>>(col, dinv, E);
    dinv_kernel<<<(n + 255) / 256, 256, 0, stream>>>(dinv, n);

    int tiles   = (n + 15) / 16;               // 6250
    int gblocks = (tiles + 7) / 8;             // 8 waves (1 tile each) per 256-thr block

    // Layer 1
    gemm_k16_n32_wmma<<<gblocks, 256, 0, stream>>>(x0, W1, h1, n);
    long long tot1 = (long long)E * 32;
    scatter_kernel<32><<<(int)((tot1 + 255) / 256), 256, 0, stream>>>(row, col, h1, dinv, agg1, E);
    selfloop_bias_relu<<<(int)(((long long)n * 32 + 255) / 256), 256, 0, stream>>>(agg1, h1, dinv, b1, n, 32);

    // Layer 2
    gemm_k32_n16_wmma<<<gblocks, 256, 0, stream>>>(agg1, W2, h2, n);
    long long tot2 = (long long)E * 16;
    scatter_kernel<16><<<(int)((tot2 + 255) / 256), 256, 0, stream>>>(row, col, h2, dinv, agg2, E);

    // Fused selfloop + bias + ReLU + FC head
    final_kernel<<<(n + 255) / 256, 256, 0, stream>>>(agg2, h2, dinv, b2, fcW, fcb, out, n);
}